// FractionalEncoding_36120674959712
// MI455X (gfx1250) — compile-verified
//
#include <hip/hip_runtime.h>
#include <cstdint>

// FractionalEncoding: out[b,n,j] = emb[b,n,j] + (j<512 ? sin(r*w[j]) : cos(r*w[j-512]))
//   r = 1/fractions[b,n],  w[k] = exp2(-k * log2(1000)/512)
// B=256, N=128, D=1024  ->  32768 rows of 1024 f32.
//
// Memory-bound streaming kernel (~256 MiB @ 23.3 TB/s ~= 11.5 us floor).
// Strategy: 1 block per row, 256 threads, float4 per lane.
//   - embedding row staged via CDNA5 GLOBAL_LOAD_ASYNC_TO_LDS_B128 (ASYNCcnt),
//     overlapped with the exp2/sin/cos computation, wave-local wait (no barrier:
//     each lane reads back exactly the LDS bytes it requested).
//   - r = 1/frac via single v_rcp_f32 (feeds ~1-ulp hardware sin/cos anyway).
//   - output written with non-temporal b128 stores (write-once stream).

typedef float v4f __attribute__((ext_vector_type(4)));

#define D_MODEL 1024
#define HALF_D  512

__global__ __launch_bounds__(256) void frac_enc_add_kernel(
    const float* __restrict__ emb,
    const float* __restrict__ fracs,
    float* __restrict__ out)
{
    __shared__ __align__(16) float smem[D_MODEL];

    const int row = blockIdx.x;          // (b*N + n)
    const int tid = threadIdx.x;         // 0..255
    const size_t rowOff = (size_t)row * D_MODEL;
    const float* __restrict__ src = emb + rowOff + tid * 4;
    float* __restrict__ dst       = out + rowOff + tid * 4;

    // ---- CDNA5 async global -> LDS staging of this lane's 16 bytes ----
    // VDST = 32-bit LDS byte address, VADDR = 64-bit global address (GV mode).
    uint32_t laddr = (uint32_t)(uintptr_t)(&smem[tid * 4]);
    uint64_t gaddr = (uint64_t)(uintptr_t)src;
    asm volatile("global_load_async_to_lds_b128 %0, %1, off"
                 :: "v"(laddr), "v"(gaddr)
                 : "memory");

    // ---- compute encoding while the async copy is in flight ----
    const float r = __builtin_amdgcn_rcpf(fracs[row]); // v_rcp_f32, block-uniform src
    const float kNeg = -9.9657842846620870436e+00f / 512.0f;  // -log2(1000)/512
    const int j0   = tid * 4;
    const int base = j0 & (HALF_D - 1);                // j mod 512

    v4f enc;
    if (j0 < HALF_D) {                                 // waves 0-3: all sin (no divergence)
        enc.x = __sinf(r * __builtin_amdgcn_exp2f(kNeg * (float)(base + 0)));
        enc.y = __sinf(r * __builtin_amdgcn_exp2f(kNeg * (float)(base + 1)));
        enc.z = __sinf(r * __builtin_amdgcn_exp2f(kNeg * (float)(base + 2)));
        enc.w = __sinf(r * __builtin_amdgcn_exp2f(kNeg * (float)(base + 3)));
    } else {                                           // waves 4-7: all cos
        enc.x = __cosf(r * __builtin_amdgcn_exp2f(kNeg * (float)(base + 0)));
        enc.y = __cosf(r * __builtin_amdgcn_exp2f(kNeg * (float)(base + 1)));
        enc.z = __cosf(r * __builtin_amdgcn_exp2f(kNeg * (float)(base + 2)));
        enc.w = __cosf(r * __builtin_amdgcn_exp2f(kNeg * (float)(base + 3)));
    }

    // ---- wave-local wait for our async loads, read back, add, stream out ----
    asm volatile("s_wait_asynccnt 0" ::: "memory");
    v4f e = *(const v4f*)(&smem[tid * 4]);             // ds_load_b128
    v4f o = e + enc;
    __builtin_nontemporal_store(o, (v4f*)dst);         // NT b128 store (write-once)
}

extern "C" void kernel_launch(void* const* d_in, const int* in_sizes, int n_in,
                              void* d_out, int out_size, void* d_ws, size_t ws_size,
                              hipStream_t stream) {
    const float* emb   = (const float*)d_in[0];   // [B, N, D] f32
    const float* fracs = (const float*)d_in[1];   // [B, N]    f32
    // d_in[2] = d_model (int scalar) -> fixed at 1024, baked into the kernel.
    float* out = (float*)d_out;                   // [B, N, D] f32

    const int rows = in_sizes[1];                 // B*N = 32768
    frac_enc_add_kernel<<<rows, 256, 0, stream>>>(emb, fracs, out);
    (void)n_in; (void)out_size; (void)d_ws; (void)ws_size;
}